// AUV_39565238730963
// MI455X (gfx1250) — compile-verified
//
#include <hip/hip_runtime.h>

#define NCH  4
#define NXD  256
#define NBAS 30
#define NF   32

typedef float v2f __attribute__((ext_vector_type(2)));
typedef float v4f __attribute__((ext_vector_type(4)));
typedef float v8f __attribute__((ext_vector_type(8)));

__device__ __forceinline__ v8f wmma4(v2f a, v2f b, v8f c) {
    // D = A(16x4 f32) * B(4x16 f32) + C(16x16 f32)
    return __builtin_amdgcn_wmma_f32_16x16x4_f32(
        false, a, false, b, (short)0, c, false, false);
}

// Per-lane constants for the 16x16 DFT matrix W16^(m*k), sliced into 4 K-chunks.
// D16 is symmetric, so the same per-lane values serve both the A-operand layout
// (M on lanes) and the B-operand layout (N on lanes).
__device__ __forceinline__ void dft16_consts(int lane, v2f dr[4], v2f di[4], v2f dni[4]) {
    const float w16 = -6.2831853071795864769f / 16.0f;
    const int m  = lane & 15;
    const int kb = ((lane >> 4) & 1) * 2;
#pragma unroll
    for (int t = 0; t < 4; ++t) {
        int K0 = 4 * t + kb;
        float s0, c0, s1, c1;
        __sincosf(w16 * (float)((m * K0) & 15), &s0, &c0);
        __sincosf(w16 * (float)((m * (K0 + 1)) & 15), &s1, &c1);
        dr[t]  = (v2f){ c0,  c1};
        di[t]  = (v2f){ s0,  s1};
        dni[t] = (v2f){-s0, -s1};
    }
}

// 256-point complex FFT of v[0..255] in place, one wave (32 lanes), as two
// radix-16 stages done with WMMA 16x16 complex matmuls + W256 twiddles.
// Final element (k1 + 16*k2) is multiplied by m1 if ((k1 + p0) odd) else m0.
__device__ __forceinline__ void fft256_wave(float2* v, int lane,
                                            const v2f dr[4], const v2f di[4], const v2f dni[4],
                                            int p0, float m0, float m1)
{
    const int n16 = lane & 15;
    const int hi  = (lane >> 4) & 1;
    const int kb  = hi * 2;

    // ---- Stage A: Y[k1][n2] = sum_{n1} W16^{n1 k1} x[16 n1 + n2] ----
    v2f xr[4], xi[4];
#pragma unroll
    for (int t = 0; t < 4; ++t) {
        int K0 = 4 * t + kb;                       // n1 for component q=0
        float2 e0 = v[16 * K0 + n16];
        float2 e1 = v[16 * (K0 + 1) + n16];
        xr[t] = (v2f){e0.x, e1.x};
        xi[t] = (v2f){e0.y, e1.y};
    }
    v8f yr = {0.f,0.f,0.f,0.f,0.f,0.f,0.f,0.f};
    v8f yi = {0.f,0.f,0.f,0.f,0.f,0.f,0.f,0.f};
#pragma unroll
    for (int t = 0; t < 4; ++t) {
        yr = wmma4(dr[t],  xr[t], yr);
        yr = wmma4(dni[t], xi[t], yr);
        yi = wmma4(di[t],  xr[t], yi);
        yi = wmma4(dr[t],  xi[t], yi);
    }
    __syncthreads();
    // C-layout -> LDS as [k1*16 + n2]
#pragma unroll
    for (int r = 0; r < 8; ++r) {
        int M = r + 8 * hi;                        // k1
        v[M * 16 + n16] = make_float2(yr[r], yi[r]);
    }
    __syncthreads();

    // ---- Twiddle + Stage C: Z[k1][k2] = sum_{n2} Y'[k1][n2] W16^{n2 k2} ----
    const float w256 = -6.2831853071795864769f / 256.0f;
    const int M = n16;                             // k1 (A-operand: M on lanes)
    v2f ar[4], ai[4];
#pragma unroll
    for (int t = 0; t < 4; ++t) {
        int K0 = 4 * t + kb;                       // n2
        float2 y0 = v[M * 16 + K0];
        float2 y1 = v[M * 16 + K0 + 1];
        float s0, c0, s1, c1;
        __sincosf(w256 * (float)((M * K0) & 255), &s0, &c0);
        __sincosf(w256 * (float)((M * (K0 + 1)) & 255), &s1, &c1);
        ar[t] = (v2f){ y0.x * c0 - y0.y * s0,  y1.x * c1 - y1.y * s1 };
        ai[t] = (v2f){ y0.x * s0 + y0.y * c0,  y1.x * s1 + y1.y * c1 };
    }
    v8f zr = {0.f,0.f,0.f,0.f,0.f,0.f,0.f,0.f};
    v8f zi = {0.f,0.f,0.f,0.f,0.f,0.f,0.f,0.f};
#pragma unroll
    for (int t = 0; t < 4; ++t) {
        zr = wmma4(ar[t], dr[t],  zr);
        zr = wmma4(ai[t], dni[t], zr);
        zi = wmma4(ar[t], di[t],  zi);
        zi = wmma4(ai[t], dr[t],  zi);
    }
    __syncthreads();
    // X[k1 + 16*k2]  (k1 = r + 8*hi, k2 = n16), with output sign/scale
#pragma unroll
    for (int r = 0; r < 8; ++r) {
        int Mz = r + 8 * hi;
        float mult = ((Mz + p0) & 1) ? m1 : m0;
        v[Mz + 16 * n16] = make_float2(zr[r] * mult, zi[r] * mult);
    }
    __syncthreads();
}

// Pass 1: coil multiply + input (-1)^(i+j) + row FFTs (j -> v). Fully coalesced.
__global__ __launch_bounds__(256) void AUV_fft_rows(const float* __restrict__ x,
                                                    const float* __restrict__ csm,
                                                    float2* __restrict__ ws)
{
    __shared__ float2 smem[8 * 256];
    const int lane = threadIdx.x & 31;
    const int wv   = threadIdx.x >> 5;
    float2* v = smem + wv * 256;

    const int vec = blockIdx.x * 8 + wv;           // [0, 120*256)
    const int row = vec & 255;
    const int m   = vec >> 8;                      // c*NBAS + b
    const int b = m % NBAS, c = m / NBAS;

    const float2* xb = (const float2*)x   + (size_t)b * 65536 + (size_t)row * 256;
    const float2* cs = (const float2*)csm + (size_t)c * 65536 + (size_t)row * 256;

#pragma unroll
    for (int e = 0; e < 8; ++e) {
        int j = e * 32 + lane;
        float2 a = xb[j];
        float2 s = cs[j];
        float pr = a.x * s.x - a.y * s.y;
        float pi = a.x * s.y + a.y * s.x;
        float sg = ((row + j) & 1) ? -1.f : 1.f;
        v[j] = make_float2(pr * sg, pi * sg);
    }
    __syncthreads();

    v2f dr[4], di[4], dni[4];
    dft16_consts(lane, dr, di, dni);
    fft256_wave(v, lane, dr, di, dni, /*p0=*/0, 1.f, 1.f);

    float2* outp = ws + (size_t)m * 65536 + (size_t)row * 256;
#pragma unroll
    for (int e = 0; e < 8; ++e) {
        int j = e * 32 + lane;
        outp[j] = v[j];
    }
}

// Pass 2: column FFTs (i -> u) + output (-1)^(u+v)/256. Each block processes an
// 8-column x 256-row panel of one image: cooperative 64B-segment global access,
// padded LDS pitch (260) keeps both the panel transpose and the FFT conflict-free.
__global__ __launch_bounds__(256) void AUV_fft_cols(float2* __restrict__ ws)
{
    __shared__ float2 smem[8 * 260];
    const int lane = threadIdx.x & 31;
    const int wv   = threadIdx.x >> 5;

    const int vecBase = blockIdx.x * 8;
    const int m       = vecBase >> 8;
    const int colBase = vecBase & 255;

    const int cs   = threadIdx.x & 7;              // column within panel
    const int rg   = threadIdx.x >> 3;             // row group 0..31
    float2* gbase  = ws + (size_t)m * 65536 + colBase + cs;

#pragma unroll
    for (int e = 0; e < 8; ++e) {
        int i = e * 32 + rg;
        smem[cs * 260 + i] = gbase[(size_t)i * 256];
    }
    __syncthreads();

    v2f dr[4], di[4], dni[4];
    dft16_consts(lane, dr, di, dni);
    const int col = colBase + wv;
    const float sc = 1.0f / 256.0f;                // ortho norm for 256x256
    fft256_wave(smem + wv * 260, lane, dr, di, dni, /*p0=*/col & 1, sc, -sc);

#pragma unroll
    for (int e = 0; e < 8; ++e) {
        int i = e * 32 + rg;
        gbase[(size_t)i * 256] = smem[cs * 260 + i];
    }
}

// Pass 3: out[c,f,n] = sum_b kr[c,b,n] * mask[f,b,n]. Dominant stream (503 MB
// mask) read once as B128 non-temporal; out written non-temporal; kr (L2-
// resident) held in 30 v4f registers per thread and read with RT policy.
__global__ __launch_bounds__(256) void AUV_einsum(const v4f* __restrict__ kr,
                                                  const v4f* __restrict__ mask,
                                                  v4f* __restrict__ out)
{
    const int p = blockIdx.x * blockDim.x + threadIdx.x;   // [0, 32768) v4f index
    const int c = blockIdx.y;                              // [0, NCH)

    const v4f* krc = kr + (size_t)c * (NBAS * 32768);
    v4f kv[NBAS];
#pragma unroll
    for (int b = 0; b < NBAS; ++b) kv[b] = krc[(size_t)b * 32768 + p];

    v4f* o = out + (size_t)c * (NF * 32768) + p;
    for (int f = 0; f < NF; ++f) {
        const v4f* mf = mask + (size_t)f * (NBAS * 32768) + p;
        v4f acc = {0.f, 0.f, 0.f, 0.f};
#pragma unroll
        for (int b = 0; b < NBAS; ++b) {
            v4f mv = __builtin_nontemporal_load(&mf[(size_t)b * 32768]);
            acc += kv[b] * mv;                      // componentwise fma
        }
        __builtin_nontemporal_store(acc, &o[(size_t)f * 32768]);
    }
}

extern "C" void kernel_launch(void* const* d_in, const int* in_sizes, int n_in,
                              void* d_out, int out_size, void* d_ws, size_t ws_size,
                              hipStream_t stream) {
    (void)in_sizes; (void)n_in; (void)out_size; (void)ws_size;
    const float* x    = (const float*)d_in[0];   // (30, 131072)
    const float* mask = (const float*)d_in[1];   // (32, 30, 131072)
    const float* csm  = (const float*)d_in[2];   // (4, 256, 256, 2)
    float2* ws2  = (float2*)d_ws;                // needs 120*65536*8 B ~= 60 MB

    const int nvec = NCH * NBAS * NXD;           // 30720 FFT vectors per pass
    dim3 blk(256);
    AUV_fft_rows<<<dim3(nvec / 8), blk, 0, stream>>>(x, csm, ws2);
    AUV_fft_cols<<<dim3(nvec / 8), blk, 0, stream>>>(ws2);
    AUV_einsum<<<dim3(32768 / 256, NCH), blk, 0, stream>>>((const v4f*)ws2,
                                                           (const v4f*)mask,
                                                           (v4f*)d_out);
}